// LocallyConnectedXYZLayer_modifyV2_10179072491792
// MI455X (gfx1250) — compile-verified
//
#include <hip/hip_runtime.h>

typedef __attribute__((ext_vector_type(2))) float v2f;
typedef float v2f_u __attribute__((ext_vector_type(2), aligned(4)));   // 4-byte aligned float2 loads
typedef __attribute__((ext_vector_type(8))) float v8f;

#define H_IMG 64
#define W_IMG 2048
#define NCLS  20
#define LDS_STRIDE 144   // 128-px block + 3 halo left + 5 halo right (K window), padded
#define LDS_ROWS   7

// out[n,c,h,x] = sum_{di,dj in 7x7} exp(-(r[h+di-3, x+dj-3]-r[h,x])^2/2) * prob[n,c,...]*mask[...]
// Banded-matmul formulation on V_WMMA_F32_16X16X4_F32:
//   D[c][x] += A[c][k] * B[k][x],  k = 24-wide input-column window per kernel row.
// A[c][k]  = prob at (class c, row h+di-3, col x0-3+k)        (loaded once per tile)
// B[k][x]  = exp(-(r_in[k]-r_out[x])^2/2)*mask[k], banded 0<=k-x<=6, 0 outside / OOB.

template<bool INTERIOR>
__device__ __forceinline__ void accumulate_rows(
    int n, int h, int x0, int xl, int hi, int w16, float r_out,
    const float* __restrict__ prob,
    const float* __restrict__ ldsR, const float* __restrict__ ldsM,
    int c_lo, int c_hi, v8f& acc0, v8f& acc1)
{
    for (int di = 0; di < 7; ++di) {
        int row_in = h + di - 3;
        if (row_in < 0 || row_in >= H_IMG) continue;   // zero-pad rows: whole B==0, skip
        const float* prow0 = prob + ((size_t)(n * NCLS + c_lo) * H_IMG + row_in) * W_IMG;
        const float* prow1 = prob + ((size_t)(n * NCLS + c_hi) * H_IMG + row_in) * W_IMG;
        const float* ldsRr = ldsR + di * LDS_STRIDE + w16;
        const float* ldsMr = ldsM + di * LDS_STRIDE + w16;
        #pragma unroll
        for (int s = 0; s < 6; ++s) {
            const int kb = 4 * s;
            const int k2 = kb + 2 * hi;   // this lane's two K values: k2, k2+1

            // ---- B fragment: banded Gaussian weights, B[k][x], x = xl ----
            float w0, w1;
            {
                int k = k2;
                int col = x0 - 3 + k;
                float d = ldsRr[k] - r_out;
                float g = __expf(-0.5f * d * d) * ldsMr[k];
                bool ok = ((unsigned)(k - xl) <= 6u) &&
                          (INTERIOR || ((unsigned)col < (unsigned)W_IMG));
                w0 = ok ? g : 0.0f;
            }
            {
                int k = k2 + 1;
                int col = x0 - 3 + k;
                float d = ldsRr[k] - r_out;
                float g = __expf(-0.5f * d * d) * ldsMr[k];
                bool ok = ((unsigned)(k - xl) <= 6u) &&
                          (INTERIOR || ((unsigned)col < (unsigned)W_IMG));
                w1 = ok ? g : 0.0f;
            }
            v2f bfrag; bfrag.x = w0; bfrag.y = w1;

            // ---- A fragments: prob at cols k2, k2+1 ----
            v2f a0, a1;
            if (INTERIOR) {
                // all cols provably in range: single unaligned float2 (global_load_b64)
                a0 = *(const v2f_u*)(prow0 + (x0 - 3 + k2));
                a1 = *(const v2f_u*)(prow1 + (x0 - 3 + k2));
            } else {
                // edge tiles: per-element clamp; garbage only where B==0
                int colA0 = min(max(x0 - 3 + k2, 0), W_IMG - 1);
                int colA1 = min(max(x0 - 2 + k2, 0), W_IMG - 1);
                a0.x = prow0[colA0]; a0.y = prow0[colA1];
                a1.x = prow1[colA0]; a1.y = prow1[colA1];
            }

            // 8-arg form: (neg_a, A, neg_b, B, c_mod, C, reuse_a, reuse_b)
            acc0 = __builtin_amdgcn_wmma_f32_16x16x4_f32(
                false, a0, false, bfrag, (short)0, acc0, false, false);
            acc1 = __builtin_amdgcn_wmma_f32_16x16x4_f32(
                false, a1, false, bfrag, (short)0, acc1, false, false);
        }
    }
}

__global__ __launch_bounds__(256) void bilateral_wmma_kernel(
    const float* __restrict__ range_img,
    const float* __restrict__ prob,
    const float* __restrict__ mask,
    float* __restrict__ out)
{
    __shared__ float ldsR[LDS_ROWS * LDS_STRIDE];
    __shared__ float ldsM[LDS_ROWS * LDS_STRIDE];

    const int n        = blockIdx.z;
    const int h        = blockIdx.y;
    const int x0_block = blockIdx.x * 128;

    const int tid  = threadIdx.x;
    const int lane = tid & 31;
    const int wv   = tid >> 5;       // 8 waves, one 16-px tile each
    const int hi   = lane >> 4;      // lane half (K split)
    const int xl   = lane & 15;      // N index (pixel) / class-within-tile

    // ---- cooperative staging of range & mask halo window (clamped; OOB killed by B=0) ----
    const size_t img_base = (size_t)n * H_IMG * W_IMG;
    for (int i = tid; i < LDS_ROWS * LDS_STRIDE; i += 256) {
        int r7 = i / LDS_STRIDE;
        int cL = i - r7 * LDS_STRIDE;
        int gr = min(max(h + r7 - 3, 0), H_IMG - 1);
        int gc = min(max(x0_block - 3 + cL, 0), W_IMG - 1);
        size_t idx = img_base + (size_t)gr * W_IMG + gc;
        ldsR[i] = range_img[idx];
        ldsM[i] = mask[idx];
    }
    __syncthreads();

    const int x0  = x0_block + wv * 16;
    const int w16 = wv * 16;
    // center range value for this lane's output pixel (row h, col x0+xl)
    const float r_out = ldsR[3 * LDS_STRIDE + w16 + 3 + xl];

    v8f acc0 = {0.f,0.f,0.f,0.f,0.f,0.f,0.f,0.f};   // classes 0..15
    v8f acc1 = {0.f,0.f,0.f,0.f,0.f,0.f,0.f,0.f};   // classes 16..19 (padded to 16)

    const int c_lo = xl;                      // A-matrix row (class), tile 0
    const int c_hi = min(16 + xl, NCLS - 1);  // tile 1, clamped (rows >=20 never stored)

    // interior <=> all A columns x0-3 .. x0+20 are in [0, W): only the first wave of the
    // first x-block and the last wave of the last x-block take the clamped edge path.
    const bool interior = (x0 >= 3) && (x0 + 21 <= W_IMG);
    if (interior)
        accumulate_rows<true >(n, h, x0, xl, hi, w16, r_out, prob, ldsR, ldsM, c_lo, c_hi, acc0, acc1);
    else
        accumulate_rows<false>(n, h, x0, xl, hi, w16, r_out, prob, ldsR, ldsM, c_lo, c_hi, acc0, acc1);

    // ---- store D: VGPR r of lane -> class (r + 8*hi [+16]), pixel x0+xl ----
    const size_t out_px = (size_t)h * W_IMG + (size_t)(x0 + xl);
    #pragma unroll
    for (int r = 0; r < 8; ++r) {
        int c = r + 8 * hi;
        out[(size_t)(n * NCLS + c) * H_IMG * W_IMG + out_px] = acc0[r];
    }
    #pragma unroll
    for (int r = 0; r < 8; ++r) {
        int c = 16 + r + 8 * hi;
        if (c < NCLS)
            out[(size_t)(n * NCLS + c) * H_IMG * W_IMG + out_px] = acc1[r];
    }
}

extern "C" void kernel_launch(void* const* d_in, const int* in_sizes, int n_in,
                              void* d_out, int out_size, void* d_ws, size_t ws_size,
                              hipStream_t stream) {
    const float* range_img = (const float*)d_in[0];
    // d_in[1] = xyz: present in the reference signature but unused by its math
    const float* prob = (const float*)d_in[2];
    const float* mask = (const float*)d_in[3];
    float* out = (float*)d_out;

    const int N = in_sizes[0] / (H_IMG * W_IMG);   // = 8
    dim3 grid(W_IMG / 128, H_IMG, N);
    dim3 block(256);
    hipLaunchKernelGGL(bilateral_wmma_kernel, grid, block, 0, stream,
                       range_img, prob, mask, out);
}